// HebbianSensory_23536420782582
// MI455X (gfx1250) — compile-verified
//
#include <hip/hip_runtime.h>
#include <cmath>

#define DIM     1024
#define TWO_DIM 2048
#define NROWS   16384   // 8 * 2048
#define NANCH   4096

typedef float v2f __attribute__((ext_vector_type(2)));
typedef float v8f __attribute__((ext_vector_type(8)));

// Full-precision fp32 WMMA: D(16x16) = A(16x4) * B(4x16) + C
// NEG bits must be zero for f32 per ISA; negation is done on fragments in VALU.
__device__ __forceinline__ v8f wmma4(v2f a, v2f b, v8f c) {
  return __builtin_amdgcn_wmma_f32_16x16x4_f32(
      /*neg_a=*/false, a, /*neg_b=*/false, b,
      /*c_mod=*/(short)0, c, /*reuse_a=*/false, /*reuse_b=*/false);
}

#define ASTRIDE 36   // [row][k] tiles: 36*4B = 144B row stride, 16B aligned
#define BSTRIDE 68   // [k][n] tiles:  68*4B = 272B row stride, 16B aligned

// ---------------------------------------------------------------------------
// CDNA5 async memory->LDS copy (ASYNCcnt-tracked), 16B per lane.
// LDS operand is the 32-bit LDS byte offset: the AS3->generic addrspacecast
// keeps the LDS offset in addr[31:0] (HW truncates generic addrs the same way).
// ---------------------------------------------------------------------------
__device__ __forceinline__ void async_b128(uint32_t lds_byte_off, const float* g) {
  asm volatile("global_load_async_to_lds_b128 %0, %1, off"
               :: "v"(lds_byte_off), "v"(g) : "memory");
}
__device__ __forceinline__ void wait_async0() {
  asm volatile("s_wait_asynccnt 0x0" ::: "memory");
}
__device__ __forceinline__ uint32_t lds_off(const void* p) {
  return (uint32_t)(uintptr_t)p;
}

// Stage a 64x32 tile, row-major on both sides, into LDS [row][k] (stride ASTRIDE).
// 512 x 16B chunks, 2 async b128 per thread.
__device__ __forceinline__ void stage_rm64x32(float* Ls, const float* __restrict__ G,
                                              int r0, int k0, int ld, int t) {
#pragma unroll
  for (int it = 0; it < 2; ++it) {
    int idx = it * 256 + t;
    int row = idx >> 3;          // 64 rows
    int c4  = (idx & 7) << 2;    // 8 float4 per row
    async_b128(lds_off(Ls + row * ASTRIDE + c4),
               G + (size_t)(r0 + row) * ld + k0 + c4);
  }
}
// Stage a 32x64 strip (memory holds B[k][n] row-major) into LDS [k][n] (stride BSTRIDE).
__device__ __forceinline__ void stage_rm32x64(float* Ls, const float* __restrict__ G,
                                              int n0, int k0, int ld, int t) {
#pragma unroll
  for (int it = 0; it < 2; ++it) {
    int idx = it * 256 + t;
    int kk = idx >> 4;           // 32 k rows, 16 float4 each
    int j4 = (idx & 15) << 2;
    async_b128(lds_off(Ls + kk * BSTRIDE + j4),
               G + (size_t)(k0 + kk) * ld + n0 + j4);
  }
}

// Fragment from [row][k] LDS layout (A tiles, and B tiles staged as [n][k]):
// lanes 0-15 -> {k, k+1}, lanes 16-31 -> {k+2, k+3}: one contiguous b64 load.
__device__ __forceinline__ v2f frag_rk(const float* Ls, int r, int k, int l16, int lh) {
  const float* p = Ls + (r + l16) * ASTRIDE + k + 2 * lh;
  v2f r2; r2.x = p[0]; r2.y = p[1]; return r2;
}
// Fragment from [k][n] LDS layout (kernel-4 B tiles).
__device__ __forceinline__ v2f frag_kn(const float* Ls, int tn, int k, int l16, int lh) {
  int kk = k + 2 * lh;
  v2f r;
  r.x = Ls[(kk    ) * BSTRIDE + tn + l16];
  r.y = Ls[(kk + 1) * BSTRIDE + tn + l16];
  return r;
}

// ---------------------------------------------------------------------------
// Kernel 1: z_flat = [ gwr@Wr.T - gwi@Wi.T + (br-bi) | gwi@Wr.T + gwr@Wi.T + (br+bi) ]
// 64x64 tile of both zr and zi per block; async double-buffered K pipeline.
// ---------------------------------------------------------------------------
__global__ __launch_bounds__(256) void zflat_kernel(
    const float* __restrict__ gwr, const float* __restrict__ gwi,
    const float* __restrict__ Wr,  const float* __restrict__ Wi,
    const float* __restrict__ br,  const float* __restrict__ bi,
    float* __restrict__ zf) {
  __shared__ float Ar[2][64 * ASTRIDE], Ai[2][64 * ASTRIDE];
  __shared__ float Brs[2][64 * ASTRIDE], Bis[2][64 * ASTRIDE];  // W staged [n][k]

  const int t = threadIdx.x;
  const int wave = t >> 5, lane = t & 31;
  const int l16 = lane & 15, lh = lane >> 4;
  const int bm = blockIdx.y * 64, bn = blockIdx.x * 64;
  const int tm = (wave >> 1) * 16, tn = (wave & 1) * 32;

  v8f zr0 = {}, zr1 = {}, zi0 = {}, zi1 = {};

  const int NSTEP = DIM / 32;
  // Preload strip 0 into buffer 0.
  stage_rm64x32(Ar[0], gwr, bm, 0, DIM, t);
  stage_rm64x32(Ai[0], gwi, bm, 0, DIM, t);
  stage_rm64x32(Brs[0], Wr, bn, 0, DIM, t);
  stage_rm64x32(Bis[0], Wi, bn, 0, DIM, t);

  for (int i = 0; i < NSTEP; ++i) {
    const int cur = i & 1, nxt = cur ^ 1;
    wait_async0();        // my copies into buf[cur] have landed in LDS
    __syncthreads();      // everyone's copies landed; prev compute on buf[nxt] done
    if (i + 1 < NSTEP) {
      const int k0 = (i + 1) * 32;
      stage_rm64x32(Ar[nxt], gwr, bm, k0, DIM, t);
      stage_rm64x32(Ai[nxt], gwi, bm, k0, DIM, t);
      stage_rm64x32(Brs[nxt], Wr, bn, k0, DIM, t);
      stage_rm64x32(Bis[nxt], Wi, bn, k0, DIM, t);
    }
#pragma unroll
    for (int k = 0; k < 32; k += 4) {
      v2f ar = frag_rk(Ar[cur], tm, k, l16, lh);
      v2f ai = frag_rk(Ai[cur], tm, k, l16, lh);
      v2f an = -ai;  // VALU negate (f32 WMMA NEG bits must be 0)
      v2f br0f = frag_rk(Brs[cur], tn,      k, l16, lh);
      v2f br1f = frag_rk(Brs[cur], tn + 16, k, l16, lh);
      v2f bi0f = frag_rk(Bis[cur], tn,      k, l16, lh);
      v2f bi1f = frag_rk(Bis[cur], tn + 16, k, l16, lh);
      zr0 = wmma4(ar, br0f, zr0); zr0 = wmma4(an, bi0f, zr0);
      zi0 = wmma4(ai, br0f, zi0); zi0 = wmma4(ar, bi0f, zi0);
      zr1 = wmma4(ar, br1f, zr1); zr1 = wmma4(an, bi1f, zr1);
      zi1 = wmma4(ai, br1f, zi1); zi1 = wmma4(ar, bi1f, zi1);
    }
  }

  const int col0 = bn + tn + l16;
  const int col1 = col0 + 16;
  const float br0v = br[col0], bi0v = bi[col0];
  const float br1v = br[col1], bi1v = bi[col1];
  const float biasr0 = br0v - bi0v, biasi0 = br0v + bi0v;
  const float biasr1 = br1v - bi1v, biasi1 = br1v + bi1v;
#pragma unroll
  for (int v = 0; v < 8; ++v) {
    int row = bm + tm + lh * 8 + v;
    size_t base = (size_t)row * TWO_DIM;
    zf[base + col0]       = zr0[v] + biasr0;
    zf[base + col1]       = zr1[v] + biasr1;
    zf[base + DIM + col0] = zi0[v] + biasi0;
    zf[base + DIM + col1] = zi1[v] + biasi1;
  }
}

// ---------------------------------------------------------------------------
// Kernel 2: logits C(16384x4096) = z_flat(16384x2048) @ palette.T
// ---------------------------------------------------------------------------
__global__ __launch_bounds__(256) void logits_kernel(
    const float* __restrict__ A, const float* __restrict__ P,
    float* __restrict__ C) {
  __shared__ float As[2][64 * ASTRIDE];
  __shared__ float Bs[2][64 * ASTRIDE];  // palette staged [a][k]

  const int t = threadIdx.x;
  const int wave = t >> 5, lane = t & 31;
  const int l16 = lane & 15, lh = lane >> 4;
  const int bm = blockIdx.y * 64, bn = blockIdx.x * 64;
  const int tm = (wave >> 1) * 16, tn = (wave & 1) * 32;

  v8f c0 = {}, c1 = {};

  const int NSTEP = TWO_DIM / 32;
  stage_rm64x32(As[0], A, bm, 0, TWO_DIM, t);
  stage_rm64x32(Bs[0], P, bn, 0, TWO_DIM, t);

  for (int i = 0; i < NSTEP; ++i) {
    const int cur = i & 1, nxt = cur ^ 1;
    wait_async0();
    __syncthreads();
    if (i + 1 < NSTEP) {
      const int k0 = (i + 1) * 32;
      stage_rm64x32(As[nxt], A, bm, k0, TWO_DIM, t);
      stage_rm64x32(Bs[nxt], P, bn, k0, TWO_DIM, t);
    }
#pragma unroll
    for (int k = 0; k < 32; k += 4) {
      v2f a  = frag_rk(As[cur], tm, k, l16, lh);
      v2f b0 = frag_rk(Bs[cur], tn,      k, l16, lh);
      v2f b1 = frag_rk(Bs[cur], tn + 16, k, l16, lh);
      c0 = wmma4(a, b0, c0);
      c1 = wmma4(a, b1, c1);
    }
  }

  const int col0 = bn + tn + l16;
#pragma unroll
  for (int v = 0; v < 8; ++v) {
    int row = bm + tm + lh * 8 + v;
    size_t base = (size_t)row * NANCH;
    C[base + col0]      = c0[v];
    C[base + col0 + 16] = c1[v];
  }
}

// ---------------------------------------------------------------------------
// Kernel 3: in-place row softmax over 4096 columns. One block per row.
// ---------------------------------------------------------------------------
__global__ __launch_bounds__(256) void softmax_kernel(float* __restrict__ buf) {
  const int row = blockIdx.x;
  float* p = buf + (size_t)row * NANCH;
  const int t = threadIdx.x;
  const int wave = t >> 5, lane = t & 31;

  float vals[16];
  float m = -INFINITY;
#pragma unroll
  for (int i = 0; i < 16; ++i) {
    vals[i] = p[t + i * 256];
    m = fmaxf(m, vals[i]);
  }
#pragma unroll
  for (int off = 16; off >= 1; off >>= 1)
    m = fmaxf(m, __shfl_xor(m, off, 32));

  __shared__ float sred[8];
  if (lane == 0) sred[wave] = m;
  __syncthreads();
  float gm = sred[0];
#pragma unroll
  for (int w = 1; w < 8; ++w) gm = fmaxf(gm, sred[w]);
  __syncthreads();

  float s = 0.0f;
#pragma unroll
  for (int i = 0; i < 16; ++i) {
    vals[i] = __expf(vals[i] - gm);
    s += vals[i];
  }
#pragma unroll
  for (int off = 16; off >= 1; off >>= 1)
    s += __shfl_xor(s, off, 32);
  if (lane == 0) sred[wave] = s;
  __syncthreads();
  float gs = 0.0f;
#pragma unroll
  for (int w = 0; w < 8; ++w) gs += sred[w];
  const float inv = 1.0f / gs;
#pragma unroll
  for (int i = 0; i < 16; ++i) p[t + i * 256] = vals[i] * inv;
}

// ---------------------------------------------------------------------------
// Kernel 4: exp_real = vis_attn@VP[:, :D] - aud_attn@AP[:, D:]
//           exp_imag = vis_attn@VP[:, D:] + aud_attn@AP[:, :D]
// ---------------------------------------------------------------------------
__global__ __launch_bounds__(256) void combine_kernel(
    const float* __restrict__ vis_attn, const float* __restrict__ aud_attn,
    const float* __restrict__ VP, const float* __restrict__ AP,
    float* __restrict__ exp_real, float* __restrict__ exp_imag) {
  __shared__ float Av[2][64 * ASTRIDE], Aa[2][64 * ASTRIDE];
  __shared__ float Bvl[2][32 * BSTRIDE], Bvh[2][32 * BSTRIDE];
  __shared__ float Bal[2][32 * BSTRIDE], Bah[2][32 * BSTRIDE];

  const int t = threadIdx.x;
  const int wave = t >> 5, lane = t & 31;
  const int l16 = lane & 15, lh = lane >> 4;
  const int bm = blockIdx.y * 64, bn = blockIdx.x * 64;  // bn over DIM (1024)
  const int tm = (wave >> 1) * 16, tn = (wave & 1) * 32;

  v8f re0 = {}, re1 = {}, im0 = {}, im1 = {};

  const int NSTEP = NANCH / 32;
  stage_rm64x32(Av[0], vis_attn, bm, 0, NANCH, t);
  stage_rm64x32(Aa[0], aud_attn, bm, 0, NANCH, t);
  stage_rm32x64(Bvl[0], VP, bn,       0, TWO_DIM, t);
  stage_rm32x64(Bvh[0], VP, bn + DIM, 0, TWO_DIM, t);
  stage_rm32x64(Bal[0], AP, bn,       0, TWO_DIM, t);
  stage_rm32x64(Bah[0], AP, bn + DIM, 0, TWO_DIM, t);

  for (int i = 0; i < NSTEP; ++i) {
    const int cur = i & 1, nxt = cur ^ 1;
    wait_async0();
    __syncthreads();
    if (i + 1 < NSTEP) {
      const int k0 = (i + 1) * 32;
      stage_rm64x32(Av[nxt], vis_attn, bm, k0, NANCH, t);
      stage_rm64x32(Aa[nxt], aud_attn, bm, k0, NANCH, t);
      stage_rm32x64(Bvl[nxt], VP, bn,       k0, TWO_DIM, t);
      stage_rm32x64(Bvh[nxt], VP, bn + DIM, k0, TWO_DIM, t);
      stage_rm32x64(Bal[nxt], AP, bn,       k0, TWO_DIM, t);
      stage_rm32x64(Bah[nxt], AP, bn + DIM, k0, TWO_DIM, t);
    }
#pragma unroll
    for (int k = 0; k < 32; k += 4) {
      v2f av = frag_rk(Av[cur], tm, k, l16, lh);
      v2f aa = frag_rk(Aa[cur], tm, k, l16, lh);
      v2f anv = -aa;
      v2f bvl0 = frag_kn(Bvl[cur], tn, k, l16, lh), bvl1 = frag_kn(Bvl[cur], tn + 16, k, l16, lh);
      v2f bvh0 = frag_kn(Bvh[cur], tn, k, l16, lh), bvh1 = frag_kn(Bvh[cur], tn + 16, k, l16, lh);
      v2f bal0 = frag_kn(Bal[cur], tn, k, l16, lh), bal1 = frag_kn(Bal[cur], tn + 16, k, l16, lh);
      v2f bah0 = frag_kn(Bah[cur], tn, k, l16, lh), bah1 = frag_kn(Bah[cur], tn + 16, k, l16, lh);
      re0 = wmma4(av, bvl0, re0); re0 = wmma4(anv, bah0, re0);
      im0 = wmma4(av, bvh0, im0); im0 = wmma4(aa,  bal0, im0);
      re1 = wmma4(av, bvl1, re1); re1 = wmma4(anv, bah1, re1);
      im1 = wmma4(av, bvh1, im1); im1 = wmma4(aa,  bal1, im1);
    }
  }

  const int col0 = bn + tn + l16;
#pragma unroll
  for (int v = 0; v < 8; ++v) {
    int row = bm + tm + lh * 8 + v;
    size_t base = (size_t)row * DIM;
    exp_real[base + col0]      = re0[v];
    exp_real[base + col0 + 16] = re1[v];
    exp_imag[base + col0]      = im0[v];
    exp_imag[base + col0 + 16] = im1[v];
  }
}

// ---------------------------------------------------------------------------
extern "C" void kernel_launch(void* const* d_in, const int* in_sizes, int n_in,
                              void* d_out, int out_size, void* d_ws, size_t ws_size,
                              hipStream_t stream) {
  const float* gw_real = (const float*)d_in[0];
  const float* gw_imag = (const float*)d_in[1];
  const float* Wr      = (const float*)d_in[2];
  const float* br      = (const float*)d_in[3];
  const float* Wi      = (const float*)d_in[4];
  const float* bi      = (const float*)d_in[5];
  const float* visP    = (const float*)d_in[6];
  const float* audP    = (const float*)d_in[7];

  float* out = (float*)d_out;
  float* exp_real = out;                                        // 16384*1024
  float* exp_imag = out + (size_t)NROWS * DIM;                  // 16384*1024
  float* vis_attn = out + (size_t)NROWS * DIM * 2;              // 16384*4096

  float* zf       = (float*)d_ws;                               // 16384*2048
  float* aud_attn = zf + (size_t)NROWS * TWO_DIM;               // 16384*4096

  dim3 blk(256);
  // Stage 1: z_flat
  zflat_kernel<<<dim3(DIM / 64, NROWS / 64), blk, 0, stream>>>(
      gw_real, gw_imag, Wr, Wi, br, bi, zf);
  // Stage 2: logits (vis -> d_out attn region, aud -> workspace)
  logits_kernel<<<dim3(NANCH / 64, NROWS / 64), blk, 0, stream>>>(zf, visP, vis_attn);
  logits_kernel<<<dim3(NANCH / 64, NROWS / 64), blk, 0, stream>>>(zf, audP, aud_attn);
  // Stage 3: in-place softmax
  softmax_kernel<<<dim3(NROWS), blk, 0, stream>>>(vis_attn);
  softmax_kernel<<<dim3(NROWS), blk, 0, stream>>>(aud_attn);
  // Stage 4: combine to exp_real / exp_imag
  combine_kernel<<<dim3(DIM / 64, NROWS / 64), blk, 0, stream>>>(
      vis_attn, aud_attn, visP, audP, exp_real, exp_imag);
}